// Net_48301202211072
// MI455X (gfx1250) — compile-verified
//
#include <hip/hip_runtime.h>
#include <hip/hip_bf16.h>

typedef float v2f __attribute__((ext_vector_type(2)));
typedef float v8f __attribute__((ext_vector_type(8)));

#define N_NODES 10000
#define N_EDGES 160000
#define F_IN    256
#define HID     500
#define KP1     5000
#define KP2     2500

// ---------------------------------------------------------------- utilities
__global__ void fill_zero(float* __restrict__ p, int n) {
    int i = blockIdx.x * blockDim.x + threadIdx.x;
    if (i < n) p[i] = 0.0f;
}

__global__ void vec_add(const float* __restrict__ a, const float* __restrict__ b,
                        float* __restrict__ c, int n) {
    int i = blockIdx.x * blockDim.x + threadIdx.x;
    if (i < n) c[i] = a[i] + b[i];
}

// ------------------------------------------------- edge scatter (segment_sum)
// one thread per (edge, 4-float chunk); F must be a multiple of 4 (256, 500 ok)
__global__ void scatter_edges(const float* __restrict__ x, const int* __restrict__ src,
                              const int* __restrict__ dst, const float* __restrict__ w,
                              float* __restrict__ agg, int E, int F) {
    int idx = blockIdx.x * blockDim.x + threadIdx.x;
    int fchunks = F >> 2;
    int e = idx / fchunks;
    if (e >= E) return;
    float we = w[e];
    if (we == 0.0f) return;                       // dropped edges contribute exactly 0
    int f = (idx - e * fchunks) << 2;
    const float4 v = *(const float4*)(x + (size_t)src[e] * F + f);
    float* ap = agg + (size_t)dst[e] * F + f;
    __hip_atomic_fetch_add(ap + 0, v.x * we, __ATOMIC_RELAXED, __HIP_MEMORY_SCOPE_AGENT);
    __hip_atomic_fetch_add(ap + 1, v.y * we, __ATOMIC_RELAXED, __HIP_MEMORY_SCOPE_AGENT);
    __hip_atomic_fetch_add(ap + 2, v.z * we, __ATOMIC_RELAXED, __HIP_MEMORY_SCOPE_AGENT);
    __hip_atomic_fetch_add(ap + 3, v.w * we, __ATOMIC_RELAXED, __HIP_MEMORY_SCOPE_AGENT);
}

// --------------------------------------------------- WMMA fp32 dual GEMM
// C[M,N] = relu(A1 @ B1^T + A2 @ B2^T + bias),  A*: [M,K] rowmaj, B*: [N,K] rowmaj
// one wave per 16x16 output tile, K stepped by 4 via V_WMMA_F32_16X16X4_F32
__global__ __launch_bounds__(256)
void gemm_dual_bias_relu(const float* __restrict__ A1, const float* __restrict__ B1,
                         const float* __restrict__ A2, const float* __restrict__ B2,
                         const float* __restrict__ bias, float* __restrict__ C,
                         int M, int N, int K) {
    const int lane = threadIdx.x & 31;
    const int wave = threadIdx.x >> 5;
    const int tilesN = (N + 15) >> 4;
    const int tilesM = (M + 15) >> 4;
    const int tile = blockIdx.x * 8 + wave;
    if (tile >= tilesM * tilesN) return;          // wave-uniform: EXEC stays all-ones
    const int m0 = (tile / tilesN) << 4;
    const int n0 = (tile % tilesN) << 4;

    const int half = lane >> 4;                   // 0: K+{0,1} / M rows 0-7 ; 1: K+{2,3} / rows 8-15
    const int l15  = lane & 15;
    int rowA = m0 + l15; if (rowA >= M) rowA = M - 1;   // clamp loads, mask stores
    int colB = n0 + l15; if (colB >= N) colB = N - 1;
    const float* a1p = A1 + (size_t)rowA * K + 2 * half;
    const float* a2p = A2 + (size_t)rowA * K + 2 * half;
    const float* b1p = B1 + (size_t)colB * K + 2 * half;
    const float* b2p = B2 + (size_t)colB * K + 2 * half;

    v8f acc = {};
    for (int k = 0; k < K; k += 4) {
        v2f a1 = *(const v2f*)(a1p + k);
        v2f b1 = *(const v2f*)(b1p + k);
        acc = __builtin_amdgcn_wmma_f32_16x16x4_f32(false, a1, false, b1,
                                                    (short)0, acc, false, false);
        v2f a2 = *(const v2f*)(a2p + k);
        v2f b2 = *(const v2f*)(b2p + k);
        acc = __builtin_amdgcn_wmma_f32_16x16x4_f32(false, a2, false, b2,
                                                    (short)0, acc, false, false);
    }

    const int col = n0 + l15;
    const float bc = (col < N) ? bias[col] : 0.0f;
#pragma unroll
    for (int r = 0; r < 8; ++r) {
        int row = m0 + r + 8 * half;
        float v = acc[r] + bc;
        v = v > 0.0f ? v : 0.0f;
        if (row < M && col < N) C[(size_t)row * N + col] = v;
    }
}

// ------------------------------------------------------ per-node pool scores
__global__ void node_scores(const float* __restrict__ h, const float* __restrict__ p,
                            float* __restrict__ score, int N, int F) {
    int gw = (blockIdx.x * blockDim.x + threadIdx.x) >> 5;
    int lane = threadIdx.x & 31;
    if (gw >= N) return;
    const float* row = h + (size_t)gw * F;
    float dot = 0.0f, pp = 0.0f;
    for (int f = lane; f < F; f += 32) { float pv = p[f]; dot += row[f] * pv; pp += pv * pv; }
    for (int off = 16; off; off >>= 1) {
        dot += __shfl_down(dot, off, 32);
        pp  += __shfl_down(pp,  off, 32);
    }
    if (lane == 0) score[gw] = tanhf(dot / sqrtf(pp));
}

// -------------------------------------------------- top-k selection (1 block)
__device__ __forceinline__ unsigned asckey(float s) {
    unsigned u = __float_as_uint(s);
    return (u & 0x80000000u) ? ~u : (u | 0x80000000u);   // monotone ascending key
}

__device__ unsigned blockScanExc(unsigned v, unsigned* buf, int t, unsigned* total) {
    buf[t] = v;
    __syncthreads();
    for (int off = 1; off < 1024; off <<= 1) {
        unsigned add = (t >= off) ? buf[t - off] : 0u;
        __syncthreads();
        buf[t] += add;
        __syncthreads();
    }
    unsigned inc = buf[t];
    *total = buf[1023];
    __syncthreads();
    return inc - v;
}

__global__ __launch_bounds__(1024)
void topk_select(const float* __restrict__ score, int N, int k,
                 int* __restrict__ kept, int* __restrict__ new_idx,
                 int* __restrict__ old_of_new, float* __restrict__ vals) {
    __shared__ unsigned hist[256];
    __shared__ unsigned scanbuf[1024];
    __shared__ unsigned sh_pm[3];     // prefix, mask, rank
    __shared__ unsigned sh_base[2];   // running eq / keep bases
    const int t = threadIdx.x;

    if (t == 0) { sh_pm[0] = 0u; sh_pm[1] = 0u; sh_pm[2] = (unsigned)k; }
    __syncthreads();

    // 4-pass MSB radix select of the k-th largest key
    for (int d = 3; d >= 0; --d) {
        if (t < 256) hist[t] = 0u;
        __syncthreads();
        unsigned prefix = sh_pm[0], mask = sh_pm[1];
        for (int i = t; i < N; i += 1024) {
            unsigned key = asckey(score[i]);
            if ((key & mask) == prefix)
                atomicAdd(&hist[(key >> (8 * d)) & 255u], 1u);
        }
        __syncthreads();
        if (t == 0) {
            unsigned rank = sh_pm[2], cum = 0u; int chosen = 0;
            for (int b = 255; b >= 0; --b) {
                unsigned c = hist[b];
                if (cum + c >= rank) { chosen = b; rank -= cum; break; }
                cum += c;
            }
            sh_pm[0] = prefix | ((unsigned)chosen << (8 * d));
            sh_pm[1] = mask | (255u << (8 * d));
            sh_pm[2] = rank;
        }
        __syncthreads();
    }
    const unsigned T = sh_pm[0];       // key of the k-th largest
    const unsigned r_eq = sh_pm[2];    // how many ==T to keep (by index order)

    if (t == 0) { sh_base[0] = 0u; sh_base[1] = 0u; }
    __syncthreads();

    // stable compaction: keep key>T, plus first r_eq elements with key==T
    for (int base = 0; base < N; base += 1024) {
        int i = base + t;
        unsigned key  = (i < N) ? asckey(score[i]) : 0u;
        unsigned isgt = (i < N && key > T)  ? 1u : 0u;
        unsigned iseq = (i < N && key == T) ? 1u : 0u;
        unsigned eqBase = sh_base[0], keepBase = sh_base[1];
        __syncthreads();
        unsigned eqTot, kTot;
        unsigned eqExc = blockScanExc(iseq, scanbuf, t, &eqTot);
        unsigned keep  = isgt | (iseq & ((eqBase + eqExc < r_eq) ? 1u : 0u));
        unsigned kExc  = blockScanExc(keep, scanbuf, t, &kTot);
        if (i < N) {
            kept[i] = (int)keep;
            int nj = (int)(keepBase + kExc);
            new_idx[i] = keep ? nj : 0;
            if (keep) { old_of_new[nj] = i; vals[nj] = score[i]; }
        }
        __syncthreads();
        if (t == 0) { sh_base[0] = eqBase + eqTot; sh_base[1] = keepBase + kTot; }
        __syncthreads();
    }
}

// --------------------------------------------- gather kept rows, scale by score
__global__ void gather_scale(const float* __restrict__ h, const int* __restrict__ oldidx,
                             const float* __restrict__ vals, float* __restrict__ xp,
                             int k, int F) {
    int idx = blockIdx.x * blockDim.x + threadIdx.x;
    if (idx >= k * F) return;
    int j = idx / F, f = idx - j * F;
    xp[idx] = h[(size_t)oldidx[j] * F + f] * vals[j];
}

__global__ void remap_edges(const int* __restrict__ src, const int* __restrict__ dst,
                            const float* __restrict__ w, const int* __restrict__ kept,
                            const int* __restrict__ nidx, int* __restrict__ s2,
                            int* __restrict__ d2, float* __restrict__ w2, int E) {
    int e = blockIdx.x * blockDim.x + threadIdx.x;
    if (e >= E) return;
    int s = src[e], d = dst[e];
    bool kp = kept[s] && kept[d];
    s2[e] = kp ? nidx[s] : 0;
    d2[e] = kp ? nidx[d] : 0;
    w2[e] = kp ? w[e] : 0.0f;
}

// ------------------------------------------------ readout: [max ; mean] -> 2F
__global__ __launch_bounds__(256)
void readout_kernel(const float* __restrict__ xp, int k, int F, float* __restrict__ out) {
    __shared__ float smax[256], ssum[256];
    int f = blockIdx.x, t = threadIdx.x;
    float mx = -3.402823466e38f, sm = 0.0f;
    for (int j = t; j < k; j += 256) {
        float v = xp[(size_t)j * F + f];
        mx = fmaxf(mx, v);
        sm += v;
    }
    smax[t] = mx; ssum[t] = sm;
    __syncthreads();
    for (int off = 128; off; off >>= 1) {
        if (t < off) { smax[t] = fmaxf(smax[t], smax[t + off]); ssum[t] += ssum[t + off]; }
        __syncthreads();
    }
    if (t == 0) { out[f] = smax[0]; out[F + f] = ssum[0] / (float)k; }
}

// --------------------------------------------------- MLP: warp per output row
__global__ void mlp_rowdot(const float* __restrict__ in, const float* __restrict__ W,
                           const float* __restrict__ b, float* __restrict__ out,
                           int K, int Nout, int act) {
    int gw = (blockIdx.x * blockDim.x + threadIdx.x) >> 5;
    int lane = threadIdx.x & 31;
    if (gw >= Nout) return;
    const float* wr = W + (size_t)gw * K;
    float dot = 0.0f;
    for (int kk = lane; kk < K; kk += 32) dot += in[kk] * wr[kk];
    for (int off = 16; off; off >>= 1) dot += __shfl_down(dot, off, 32);
    if (lane == 0) {
        float v = dot + b[gw];
        out[gw] = (act == 0) ? fmaxf(v, 0.0f) : 1.0f / (1.0f + expf(-v));
    }
}

// ---------------------------------------------------------------- launcher
extern "C" void kernel_launch(void* const* d_in, const int* in_sizes, int n_in,
                              void* d_out, int out_size, void* d_ws, size_t ws_size,
                              hipStream_t stream) {
    const float* x    = (const float*)d_in[0];
    const int*   esrc = (const int*)  d_in[1];
    const int*   edst = (const int*)  d_in[2];
    const float* ew   = (const float*)d_in[3];
    const float* W1r  = (const float*)d_in[4];
    const float* b1   = (const float*)d_in[5];
    const float* W1o  = (const float*)d_in[6];
    const float* p1   = (const float*)d_in[7];
    const float* W2r  = (const float*)d_in[8];
    const float* b2   = (const float*)d_in[9];
    const float* W2o  = (const float*)d_in[10];
    const float* p2   = (const float*)d_in[11];
    const float* L1W  = (const float*)d_in[12];
    const float* L1b  = (const float*)d_in[13];
    const float* L2W  = (const float*)d_in[14];
    const float* L2b  = (const float*)d_in[15];
    const float* L3W  = (const float*)d_in[16];
    const float* L3b  = (const float*)d_in[17];
    float* out = (float*)d_out;

    char* ws = (char*)d_ws;
    size_t off = 0;
    auto alloc = [&](size_t bytes) -> void* {
        void* p = ws + off;
        off = (off + bytes + 255) & ~(size_t)255;
        return p;
    };

    float* agg1   = (float*)alloc((size_t)N_NODES * F_IN * 4);
    float* h1     = (float*)alloc((size_t)N_NODES * HID * 4);
    float* score1 = (float*)alloc((size_t)N_NODES * 4);
    int*   kept1  = (int*)  alloc((size_t)N_NODES * 4);
    int*   nidx1  = (int*)  alloc((size_t)N_NODES * 4);
    int*   old1   = (int*)  alloc((size_t)KP1 * 4);
    float* vals1  = (float*)alloc((size_t)KP1 * 4);
    float* xp1    = (float*)alloc((size_t)KP1 * HID * 4);
    int*   s2     = (int*)  alloc((size_t)N_EDGES * 4);
    int*   d2     = (int*)  alloc((size_t)N_EDGES * 4);
    float* w2     = (float*)alloc((size_t)N_EDGES * 4);
    float* agg2   = (float*)alloc((size_t)KP1 * HID * 4);
    float* h2     = (float*)alloc((size_t)KP1 * HID * 4);
    float* score2 = (float*)alloc((size_t)KP1 * 4);
    int*   kept2  = (int*)  alloc((size_t)KP1 * 4);
    int*   nidx2  = (int*)  alloc((size_t)KP1 * 4);
    int*   old2   = (int*)  alloc((size_t)KP2 * 4);
    float* vals2  = (float*)alloc((size_t)KP2 * 4);
    float* xp2    = (float*)alloc((size_t)KP2 * HID * 4);
    float* xr1    = (float*)alloc((size_t)2 * HID * 4);
    float* xr2    = (float*)alloc((size_t)2 * HID * 4);
    float* z      = (float*)alloc((size_t)2 * HID * 4);
    float* m1     = (float*)alloc((size_t)2000 * 4);
    float* m2     = (float*)alloc((size_t)4000 * 4);

    const int TB = 256;
    auto cdiv = [](long long a, long long b) -> int { return (int)((a + b - 1) / b); };

    // ---- conv1: h1 = relu(segsum(x[src]*w, dst) @ W1r^T + b1 + x @ W1o^T)
    fill_zero<<<cdiv((long long)N_NODES * F_IN, TB), TB, 0, stream>>>(agg1, N_NODES * F_IN);
    scatter_edges<<<cdiv((long long)N_EDGES * (F_IN / 4), TB), TB, 0, stream>>>(
        x, esrc, edst, ew, agg1, N_EDGES, F_IN);
    {
        int tiles = ((N_NODES + 15) / 16) * ((HID + 15) / 16);
        gemm_dual_bias_relu<<<cdiv(tiles, 8), TB, 0, stream>>>(
            agg1, W1r, x, W1o, b1, h1, N_NODES, HID, F_IN);
    }

    // ---- pool1 (k = 5000)
    node_scores<<<cdiv((long long)N_NODES * 32, TB), TB, 0, stream>>>(h1, p1, score1, N_NODES, HID);
    topk_select<<<1, 1024, 0, stream>>>(score1, N_NODES, KP1, kept1, nidx1, old1, vals1);
    gather_scale<<<cdiv((long long)KP1 * HID, TB), TB, 0, stream>>>(h1, old1, vals1, xp1, KP1, HID);
    remap_edges<<<cdiv(N_EDGES, TB), TB, 0, stream>>>(esrc, edst, ew, kept1, nidx1, s2, d2, w2, N_EDGES);
    readout_kernel<<<HID, TB, 0, stream>>>(xp1, KP1, HID, xr1);

    // ---- conv2 on pooled graph
    fill_zero<<<cdiv((long long)KP1 * HID, TB), TB, 0, stream>>>(agg2, KP1 * HID);
    scatter_edges<<<cdiv((long long)N_EDGES * (HID / 4), TB), TB, 0, stream>>>(
        xp1, s2, d2, w2, agg2, N_EDGES, HID);
    {
        int tiles = ((KP1 + 15) / 16) * ((HID + 15) / 16);
        gemm_dual_bias_relu<<<cdiv(tiles, 8), TB, 0, stream>>>(
            agg2, W2r, xp1, W2o, b2, h2, KP1, HID, HID);
    }

    // ---- pool2 (k = 2500)
    node_scores<<<cdiv((long long)KP1 * 32, TB), TB, 0, stream>>>(h2, p2, score2, KP1, HID);
    topk_select<<<1, 1024, 0, stream>>>(score2, KP1, KP2, kept2, nidx2, old2, vals2);
    gather_scale<<<cdiv((long long)KP2 * HID, TB), TB, 0, stream>>>(h2, old2, vals2, xp2, KP2, HID);
    readout_kernel<<<HID, TB, 0, stream>>>(xp2, KP2, HID, xr2);

    // ---- head: z = x1 + x2 ; MLP 1000 -> 2000 -> 4000 -> 100
    vec_add<<<cdiv(2 * HID, TB), TB, 0, stream>>>(xr1, xr2, z, 2 * HID);
    mlp_rowdot<<<cdiv((long long)2000 * 32, TB), TB, 0, stream>>>(z,  L1W, L1b, m1, 1000, 2000, 0);
    mlp_rowdot<<<cdiv((long long)4000 * 32, TB), TB, 0, stream>>>(m1, L2W, L2b, m2, 2000, 4000, 0);
    mlp_rowdot<<<cdiv((long long)100  * 32, TB), TB, 0, stream>>>(m2, L3W, L3b, out, 4000, 100, 1);
}